// SelfAttentionFusion_46617575030956
// MI455X (gfx1250) — compile-verified
//
#include <hip/hip_runtime.h>
#include <stdint.h>

typedef __attribute__((ext_vector_type(2))) float v2f;
typedef __attribute__((ext_vector_type(8))) float v8f;

#define NTOT 16384
#define DV   512
#define SPW  16          // samples per workgroup
#define NCHUNK 4         // chunks of 4 samples (M=32 rows)
#define XS  516          // LDS row strides (== 4 mod 64 banks -> conflict-free column reads)
#define QKS 388          // qkv head-pair buffer stride (384 cols + pad)
#define AS  516
#define HS  260
#define RS  516
#define EPSV 1e-5f

static __device__ __forceinline__ v8f wmma4(v2f a, v2f b, v8f c) {
    return __builtin_amdgcn_wmma_f32_16x16x4_f32(false, a, false, b, (short)0, c, false, false);
}

__launch_bounds__(256, 1)
__global__ void saf_fused_kernel(
    const float* __restrict__ x,     // (N,8,512)
    const float* __restrict__ Wqkv,  // (1536,512)
    const float* __restrict__ W1,    // (256,512)
    const float* __restrict__ b1,    // (256)
    const float* __restrict__ W2,    // (1,256)
    const float* __restrict__ b2,    // (1)
    const float* __restrict__ Wo,    // (512,512)
    const float* __restrict__ bo,    // (512)
    const float* __restrict__ gamma, // (512)
    const float* __restrict__ beta,  // (512)
    float* __restrict__ out,         // (N,512)
    float* __restrict__ fw_out)      // (N,8)
{
    __shared__ float xs[32 * XS];       // x tile, 32 rows (4 samples x B=8)     66 KB
    __shared__ float qkvh[32 * QKS];    // q|k|v columns of one head-pair         50 KB
    __shared__ float attns[32 * AS];    // attn_out tile; rows 0..15 reused as residual
    __shared__ float h1s[32 * HS];      // fusion MLP hidden 32 x 256             33 KB
    __shared__ float fusedb[16 * RS];   // fused row per sample (persistent)      33 KB
    __shared__ float xmeanb[16 * RS];   // mean over B per sample (persistent)    33 KB
    __shared__ float fsc[32];
    __shared__ float wgt[32];

    const int tid  = threadIdx.x;
    const int w    = tid >> 5;      // wave id 0..7
    const int lane = tid & 31;
    const int n0   = blockIdx.x * SPW;

    const int rowA = lane & 15;          // A-frag / B-frag row-col within a 16-tile
    const int kh   = (lane >> 4) << 1;   // K sub-offset 0 or 2
    const int crow = (lane >> 4) << 3;   // C-frag row base 0 or 8

    const v8f vzero = {0.f, 0.f, 0.f, 0.f, 0.f, 0.f, 0.f, 0.f};

    #pragma unroll 1
    for (int c = 0; c < NCHUNK; ++c) {
        const int r0 = (n0 + 4 * c) * 8;   // first global x-row of this chunk (32 rows)

        // ---- Phase A: async-stage x tile (32 x 512 f32 = 64 KB) straight into LDS ----
        #pragma unroll
        for (int i = 0; i < 16; ++i) {
            int idx = tid + 256 * i;          // 0..4095 float4 slots
            int r  = idx >> 7;
            int c4 = (idx & 127) << 2;
            unsigned lds_addr = (unsigned)(uintptr_t)&xs[r * XS + c4];
            unsigned long long gaddr =
                (unsigned long long)(uintptr_t)&x[(size_t)(r0 + r) * DV + c4];
            asm volatile("global_load_async_to_lds_b128 %0, %1, off"
                         :: "v"(lds_addr), "v"(gaddr) : "memory");
        }
        asm volatile("s_wait_asynccnt 0" ::: "memory");
        __syncthreads();

        // ---- mean over B (residual input) ----
        {
            int s  = tid >> 6;                 // sample within chunk 0..3
            int d0 = (tid & 63) << 3;          // 8 floats per thread
            float m[8] = {0.f,0.f,0.f,0.f,0.f,0.f,0.f,0.f};
            #pragma unroll
            for (int b = 0; b < 8; ++b) {
                const float* xp = &xs[(s * 8 + b) * XS + d0];
                #pragma unroll
                for (int j = 0; j < 8; ++j) m[j] += xp[j];
            }
            float* mp = &xmeanb[(c * 4 + s) * RS + d0];
            #pragma unroll
            for (int j = 0; j < 8; ++j) mp[j] = m[j] * 0.125f;
        }

        // ---- Phases B+C: per head-pair, QKV GEMM (M=32, N=384, K=512) + attention ----
        #pragma unroll 1
        for (int hi = 0; hi < 4; ++hi) {
            {   // QKV GEMM: 24 column tiles / 8 waves = 3 each; each wave does both
                // 16-row M-tiles so every Wqkv fragment is loaded once per chunk.
                v8f acc[3][2];
                #pragma unroll
                for (int j = 0; j < 3; ++j) { acc[j][0] = vzero; acc[j][1] = vzero; }
                for (int k = 0; k < 512; k += 4) {
                    const float2 av0 = *(const float2*)&xs[rowA * XS + k + kh];
                    const float2 av1 = *(const float2*)&xs[(rowA + 16) * XS + k + kh];
                    v2f a0; a0.x = av0.x; a0.y = av0.y;
                    v2f a1; a1.x = av1.x; a1.y = av1.y;
                    #pragma unroll
                    for (int j = 0; j < 3; ++j) {
                        int ct  = w * 3 + j;
                        int sec = ct >> 3, rem = ct & 7;      // sec: 0=q 1=k 2=v
                        int e = sec * 512 + (hi * 2 + (rem >> 2)) * 64
                              + ((rem & 3) << 4) + rowA;      // Wqkv output row
                        const float2 bv = *(const float2*)&Wqkv[(size_t)e * DV + k + kh];
                        v2f b; b.x = bv.x; b.y = bv.y;
                        acc[j][0] = wmma4(a0, b, acc[j][0]);
                        acc[j][1] = wmma4(a1, b, acc[j][1]);
                    }
                }
                #pragma unroll
                for (int j = 0; j < 3; ++j) {
                    int ct  = w * 3 + j;
                    int sec = ct >> 3, rem = ct & 7;
                    int cc0 = sec * 128 + rem * 16;           // column base in qkvh
                    #pragma unroll
                    for (int m = 0; m < 2; ++m)
                        #pragma unroll
                        for (int r = 0; r < 8; ++r)
                            qkvh[(m * 16 + crow + r) * QKS + cc0 + rowA] = acc[j][m][r];
                }
            }
            __syncthreads();

            {   // attention: 8 units (4 samples x 2 heads) -> one per wave
                int s = w >> 1, hl = w & 1;
                int bq = lane >> 2, j4 = lane & 3;
                const float* qp  = &qkvh[(s * 8 + bq) * QKS + hl * 64];
                const float* kp0 = &qkvh[(s * 8 + j4) * QKS + 128 + hl * 64];
                const float* kp1 = &qkvh[(s * 8 + j4 + 4) * QKS + 128 + hl * 64];
                float s0 = 0.f, s1 = 0.f;
                #pragma unroll 8
                for (int d = 0; d < 64; ++d) {
                    float qv = qp[d];
                    s0 += qv * kp0[d];
                    s1 += qv * kp1[d];
                }
                s0 *= 0.125f; s1 *= 0.125f;          // 1/sqrt(HD=64)
                float m = fmaxf(s0, s1);
                m = fmaxf(m, __shfl_xor(m, 1, 32));
                m = fmaxf(m, __shfl_xor(m, 2, 32));
                float e0 = __expf(s0 - m), e1 = __expf(s1 - m);
                float sm = e0 + e1;
                sm += __shfl_xor(sm, 1, 32);
                sm += __shfl_xor(sm, 2, 32);
                float inv = 1.f / sm;
                float p0 = e0 * inv, p1 = e1 * inv;
                int base = lane & ~3;
                float pj[8];
                #pragma unroll
                for (int q = 0; q < 4; ++q) {
                    pj[q]     = __shfl(p0, base + q, 32);
                    pj[4 + q] = __shfl(p1, base + q, 32);
                }
                int dbase = j4 * 16;
                float* op = &attns[(s * 8 + bq) * AS + (hi * 2 + hl) * 64 + dbase];
                #pragma unroll 4
                for (int dd = 0; dd < 16; ++dd) {
                    float o = 0.f;
                    #pragma unroll
                    for (int bk = 0; bk < 8; ++bk)
                        o += pj[bk] * qkvh[(s * 8 + bk) * QKS + 256 + hl * 64 + dbase + dd];
                    op[dd] = o;
                }
            }
            __syncthreads();
        }

        // ---- Phase D: h1 = relu(attn @ W1^T + b1)  (32 x 256) ----
        {
            v8f acc[2][2];
            #pragma unroll
            for (int j = 0; j < 2; ++j) { acc[j][0] = vzero; acc[j][1] = vzero; }
            for (int k = 0; k < 512; k += 4) {
                const float2 av0 = *(const float2*)&attns[rowA * AS + k + kh];
                const float2 av1 = *(const float2*)&attns[(rowA + 16) * AS + k + kh];
                v2f a0; a0.x = av0.x; a0.y = av0.y;
                v2f a1; a1.x = av1.x; a1.y = av1.y;
                #pragma unroll
                for (int j = 0; j < 2; ++j) {
                    int e = ((w * 2 + j) << 4) + rowA;
                    const float2 bv = *(const float2*)&W1[(size_t)e * DV + k + kh];
                    v2f b; b.x = bv.x; b.y = bv.y;
                    acc[j][0] = wmma4(a0, b, acc[j][0]);
                    acc[j][1] = wmma4(a1, b, acc[j][1]);
                }
            }
            #pragma unroll
            for (int j = 0; j < 2; ++j) {
                int col = ((w * 2 + j) << 4) + rowA;
                float bias = b1[col];
                #pragma unroll
                for (int m = 0; m < 2; ++m)
                    #pragma unroll
                    for (int r = 0; r < 8; ++r) {
                        float v = acc[j][m][r] + bias;
                        h1s[(m * 16 + crow + r) * HS + col] = fmaxf(v, 0.f);
                    }
            }
        }
        __syncthreads();

        // ---- Phase E: fusion scores, softmax over B, weighted pooling ----
        {
            int l16 = lane & 15;
            #pragma unroll
            for (int i = 0; i < 2; ++i) {
                int rowE = w * 4 + ((lane >> 4) << 1) + i;
                float p = 0.f;
                #pragma unroll
                for (int jj = 0; jj < 16; ++jj) {
                    int cc = l16 + jj * 16;
                    p += h1s[rowE * HS + cc] * W2[cc];
                }
                p += __shfl_xor(p, 1, 32);
                p += __shfl_xor(p, 2, 32);
                p += __shfl_xor(p, 4, 32);
                p += __shfl_xor(p, 8, 32);
                if (l16 == 0) fsc[rowE] = p + b2[0];
            }
        }
        __syncthreads();
        if (tid < 4) {
            int s = tid;
            float m = fsc[s * 8];
            #pragma unroll
            for (int b = 1; b < 8; ++b) m = fmaxf(m, fsc[s * 8 + b]);
            float e[8]; float sum = 0.f;
            #pragma unroll
            for (int b = 0; b < 8; ++b) { e[b] = __expf(fsc[s * 8 + b] - m); sum += e[b]; }
            float inv = 1.f / sum;
            #pragma unroll
            for (int b = 0; b < 8; ++b) {
                float wv = e[b] * inv;
                wgt[s * 8 + b] = wv;
                fw_out[(size_t)(n0 + c * 4 + s) * 8 + b] = wv;
            }
        }
        __syncthreads();
        {
            int s  = tid >> 6;
            int d0 = (tid & 63) << 3;
            float o[8] = {0.f,0.f,0.f,0.f,0.f,0.f,0.f,0.f};
            #pragma unroll
            for (int b = 0; b < 8; ++b) {
                float wv = wgt[s * 8 + b];
                const float* ap = &attns[(s * 8 + b) * AS + d0];
                #pragma unroll
                for (int j = 0; j < 8; ++j) o[j] += wv * ap[j];
            }
            float* fp = &fusedb[(c * 4 + s) * RS + d0];
            #pragma unroll
            for (int j = 0; j < 8; ++j) fp[j] = o[j];
        }
        __syncthreads();
    } // chunk loop

    // ---- Phase F: out = fused @ Wo^T + bo + xmean  (16 x 512), into attns rows 0..15 ----
    {
        v8f acc[4];
        #pragma unroll
        for (int t = 0; t < 4; ++t) acc[t] = vzero;
        for (int k = 0; k < 512; k += 4) {
            const float2 av = *(const float2*)&fusedb[rowA * RS + k + kh];
            v2f a; a.x = av.x; a.y = av.y;
            #pragma unroll
            for (int t = 0; t < 4; ++t) {
                int e = ((w * 4 + t) << 4) + rowA;
                const float2 bv = *(const float2*)&Wo[(size_t)e * DV + k + kh];
                v2f b; b.x = bv.x; b.y = bv.y;
                acc[t] = wmma4(a, b, acc[t]);
            }
        }
        #pragma unroll
        for (int t = 0; t < 4; ++t) {
            int e0 = (w * 4 + t) << 4;
            #pragma unroll
            for (int r = 0; r < 8; ++r) {
                int row = crow + r, col = e0 + rowA;
                attns[row * AS + col] = acc[t][r] + bo[col] + xmeanb[row * RS + col];
            }
        }
    }
    __syncthreads();

    // ---- Phase G: LayerNorm per sample row, write output ----
    {
        int row = w * 2 + (lane >> 4);
        int l16 = lane & 15;
        float sum = 0.f, sq = 0.f;
        #pragma unroll 8
        for (int j = 0; j < 32; ++j) {
            float v = attns[row * AS + j * 16 + l16];
            sum += v; sq += v * v;
        }
        sum += __shfl_xor(sum, 1, 32); sq += __shfl_xor(sq, 1, 32);
        sum += __shfl_xor(sum, 2, 32); sq += __shfl_xor(sq, 2, 32);
        sum += __shfl_xor(sum, 4, 32); sq += __shfl_xor(sq, 4, 32);
        sum += __shfl_xor(sum, 8, 32); sq += __shfl_xor(sq, 8, 32);
        float mu  = sum * (1.f / 512.f);
        float var = sq * (1.f / 512.f) - mu * mu;
        float rstd = rsqrtf(var + EPSV);
        size_t gbase = (size_t)(n0 + row) * DV;
        #pragma unroll 8
        for (int j = 0; j < 32; ++j) {
            int d = j * 16 + l16;
            float v = (attns[row * AS + d] - mu) * rstd * gamma[d] + beta[d];
            out[gbase + d] = v;
        }
    }
}

extern "C" void kernel_launch(void* const* d_in, const int* in_sizes, int n_in,
                              void* d_out, int out_size, void* d_ws, size_t ws_size,
                              hipStream_t stream) {
    (void)in_sizes; (void)n_in; (void)out_size; (void)d_ws; (void)ws_size;
    const float* x     = (const float*)d_in[0];
    const float* Wqkv  = (const float*)d_in[1];
    const float* W1    = (const float*)d_in[2];
    const float* b1    = (const float*)d_in[3];
    const float* W2    = (const float*)d_in[4];
    const float* b2    = (const float*)d_in[5];
    const float* Wo    = (const float*)d_in[6];
    const float* bo    = (const float*)d_in[7];
    const float* gamma = (const float*)d_in[8];
    const float* beta  = (const float*)d_in[9];
    float* outp = (float*)d_out;                       // (N,512) first
    float* fwp  = outp + (size_t)NTOT * DV;            // then fusion_weights (N,8)

    dim3 grid(NTOT / SPW);   // 1024 workgroups
    dim3 block(256);         // 8 wave32 waves
    saf_fused_kernel<<<grid, block, 0, stream>>>(
        x, Wqkv, W1, b1, W2, b2, Wo, bo, gamma, beta, outp, fwp);
}